// RefCondMul_65472481460821
// MI455X (gfx1250) — compile-verified
//
#include <hip/hip_runtime.h>

// Problem constants (from reference)
#define CLASSES 1000
#define MDIM    64        // input features (K of the GEMM)
#define NDIM    64        // output features
#define BDIM    4
#define LDIM    8192
#define NSAMP   (BDIM * LDIM)   // 32768

// LDS padding to avoid bank conflicts (64 banks x 4B)
#define A_STRIDE 68       // floats per A row
#define W_STRIDE 80       // floats per W row

typedef float v2f __attribute__((ext_vector_type(2)));
typedef float v8f __attribute__((ext_vector_type(8)));

// ---------------- sort-by-class pipeline ----------------

__global__ __launch_bounds__(256) void zero_kernel(int* __restrict__ counts) {
  int i = blockIdx.x * blockDim.x + threadIdx.x;
  if (i < CLASSES) counts[i] = 0;
}

__global__ __launch_bounds__(256) void hist_kernel(const long long* __restrict__ inds,
                                                   int* __restrict__ counts) {
  int n = blockIdx.x * blockDim.x + threadIdx.x;
  if (n < NSAMP) atomicAdd(&counts[(int)inds[n]], 1);
}

// single-block exclusive scan over CLASSES entries (Hillis-Steele in LDS)
__global__ __launch_bounds__(1024) void scan_kernel(const int* __restrict__ counts,
                                                    int* __restrict__ offsets,
                                                    int* __restrict__ cursor) {
  __shared__ int buf[1024];
  int t = threadIdx.x;
  int v = (t < CLASSES) ? counts[t] : 0;
  buf[t] = v;
  __syncthreads();
  for (int off = 1; off < 1024; off <<= 1) {
    int tmp = 0;
    if (t >= off) tmp = buf[t - off];
    __syncthreads();
    if (t >= off) buf[t] += tmp;
    __syncthreads();
  }
  int incl = buf[t];
  int excl = incl - v;
  if (t < CLASSES) { offsets[t] = excl; cursor[t] = excl; }
  if (t == CLASSES - 1) offsets[CLASSES] = incl;  // total
}

__global__ __launch_bounds__(256) void scatter_kernel(const long long* __restrict__ inds,
                                                      int* __restrict__ cursor,
                                                      int* __restrict__ sorted) {
  int n = blockIdx.x * blockDim.x + threadIdx.x;
  if (n < NSAMP) {
    int c = (int)inds[n];
    int p = atomicAdd(&cursor[c], 1);
    sorted[p] = n;
  }
}

// ---------------- per-class batched GEMM with f32 WMMA ----------------
// One block per class, 128 threads = 4 wave32. Each wave owns a 16-wide
// output column tile. Per 16-sample chunk: D(16x64) = A(16x64) @ W(64x64) + bias.
__global__ __launch_bounds__(128) void gemm_kernel(const float* __restrict__ x,
                                                   const float* __restrict__ w,
                                                   const float* __restrict__ bias,
                                                   const int* __restrict__ offsets,
                                                   const int* __restrict__ sorted,
                                                   float* __restrict__ out) {
  __shared__ float Wlds[MDIM * W_STRIDE];   // 20 KB
  __shared__ float Alds[16 * A_STRIDE];     // ~4.25 KB
  __shared__ int   Srow[16];

  const int c   = blockIdx.x;
  const int beg = offsets[c];
  const int cnt = offsets[c + 1] - beg;
  if (cnt == 0) return;                     // block-uniform

  const int tid  = threadIdx.x;
  const int wave = tid >> 5;                // 0..3 -> output col tile
  const int lane = tid & 31;
  const int half = lane >> 4;               // 0: K+0/K+1 rows, 1: K+2/K+3 rows
  const int ln   = lane & 15;

  // Stage W_c into LDS: 4096 floats, float4 x 8 per thread
  const float* wc = w + (size_t)c * (MDIM * NDIM);
  for (int i = tid; i < (MDIM * NDIM) / 4; i += 128) {
    float4 v = ((const float4*)wc)[i];
    int k = (i * 4) >> 6;
    int o = (i * 4) & 63;
    float* dst = &Wlds[k * W_STRIDE + o];
    dst[0] = v.x; dst[1] = v.y; dst[2] = v.z; dst[3] = v.w;
  }
  // bias value for this lane's output column (replicated over all 8 C rows)
  const float bval = bias[(size_t)c * NDIM + wave * 16 + ln];

  for (int chunk = 0; chunk < cnt; chunk += 16) {
    const int rows = min(16, cnt - chunk);
    __syncthreads();                         // Wlds ready / prev chunk done
    if (tid < 16) Srow[tid] = (tid < rows) ? sorted[beg + chunk + tid] : -1;
    __syncthreads();

    // Stage A tile (16 samples x 64 K) from strided x into LDS
    {
      const int row = tid >> 3;              // 0..15
      const int k0  = (tid & 7) * 8;         // 8 consecutive K per thread
      const int sid = Srow[row];
      if (sid >= 0) {
        const int bI = sid >> 13;            // / LDIM
        const int lI = sid & (LDIM - 1);
        const float* xp = x + ((size_t)bI * MDIM) * LDIM + lI;
        #pragma unroll
        for (int kk = 0; kk < 8; ++kk)
          Alds[row * A_STRIDE + k0 + kk] = xp[(size_t)(k0 + kk) * LDIM];
      } else {
        #pragma unroll
        for (int kk = 0; kk < 8; ++kk)
          Alds[row * A_STRIDE + k0 + kk] = 0.0f;
      }
    }
    __syncthreads();

    // C init = bias broadcast (C/D layout: VGPR r -> M=r (lanes 0-15), M=r+8 (16-31))
    v8f acc;
    #pragma unroll
    for (int r = 0; r < 8; ++r) acc[r] = bval;

    // K loop: 16 x v_wmma_f32_16x16x4_f32, fully unrolled
    #pragma unroll
    for (int kk = 0; kk < MDIM; kk += 4) {
      const int ka = kk + half * 2;
      // A frag (16x4 f32): lane holds M=ln, K=ka..ka+1
      v2f afrag = *(const v2f*)&Alds[ln * A_STRIDE + ka];
      // B frag (4x16 f32): lane holds N=wave*16+ln, K=ka..ka+1
      v2f bfrag;
      bfrag.x = Wlds[(ka + 0) * W_STRIDE + wave * 16 + ln];
      bfrag.y = Wlds[(ka + 1) * W_STRIDE + wave * 16 + ln];
      acc = __builtin_amdgcn_wmma_f32_16x16x4_f32(
          /*neg_a=*/false, afrag, /*neg_b=*/false, bfrag,
          /*c_mod=*/(short)0, acc, /*reuse_a=*/false, /*reuse_b=*/false);
    }

    // Scatter D to out[B, NDIM, LDIM]
    #pragma unroll
    for (int r = 0; r < 8; ++r) {
      const int row = r + half * 8;
      const int sid = Srow[row];
      if (sid >= 0) {
        const int bI = sid >> 13;
        const int lI = sid & (LDIM - 1);
        const int o  = wave * 16 + ln;
        out[((size_t)bI * NDIM + o) * LDIM + lI] = acc[r];
      }
    }
  }
}

extern "C" void kernel_launch(void* const* d_in, const int* in_sizes, int n_in,
                              void* d_out, int out_size, void* d_ws, size_t ws_size,
                              hipStream_t stream) {
  const float*     x    = (const float*)d_in[0];
  const long long* inds = (const long long*)d_in[1];   // jnp.int64
  const float*     w    = (const float*)d_in[2];
  const float*     bias = (const float*)d_in[3];
  float*           out  = (float*)d_out;

  // Scratch layout (ints): counts | offsets(+1) | cursor | sorted ids
  int* wsI     = (int*)d_ws;
  int* counts  = wsI;
  int* offsets = wsI + 1024;
  int* cursor  = wsI + 2176;
  int* sorted  = wsI + 3328;

  zero_kernel   <<<(CLASSES + 255) / 256, 256, 0, stream>>>(counts);
  hist_kernel   <<<(NSAMP   + 255) / 256, 256, 0, stream>>>(inds, counts);
  scan_kernel   <<<1, 1024, 0, stream>>>(counts, offsets, cursor);
  scatter_kernel<<<(NSAMP   + 255) / 256, 256, 0, stream>>>(inds, cursor, sorted);
  gemm_kernel   <<<CLASSES, 128, 0, stream>>>(x, w, bias, offsets, sorted, out);
}